// DotProductAttention_59081570125023
// MI455X (gfx1250) — compile-verified
//
#include <hip/hip_runtime.h>

// CDNA5 / gfx1250: wave32, WMMA. Memory-bound batched micro-attention:
// per batch b: Q[32,16], V[32,16] -> score = Q V^T [32,32] -> softmax rows ->
// context = attn V [32,16]. One wave per batch, fp32 WMMA 16x16x4 for both GEMMs.
// Softmax done row-per-lane through LDS (no cross-lane shuffles).

typedef __attribute__((ext_vector_type(2))) float v2f;
typedef __attribute__((ext_vector_type(4))) float v4f;
typedef __attribute__((ext_vector_type(8))) float v8f;

constexpr int SQ = 32;   // sequence dim
constexpr int DH = 16;   // head dim
constexpr int VPAD = 20; // LDS row pitch for V  (conflict-free strided reads, 16B-aligned f4 stores)
constexpr int APAD = 36; // LDS row pitch for attn (conflict-free strided v2f reads, 16B-aligned rows)
constexpr int WPB = 8;   // waves (batches) per block

__global__ __launch_bounds__(WPB * 32) void
dpa_fused_kernel(const float* __restrict__ q_g, const float* __restrict__ v_g,
                 float* __restrict__ ctx_out, float* __restrict__ attn_out) {
    __shared__ float lds_v[WPB][SQ * VPAD];
    __shared__ float lds_a[WPB][SQ * APAD];

    const int lane  = threadIdx.x & 31;
    const int wave  = threadIdx.x >> 5;
    const int batch = blockIdx.x * WPB + wave;

    const int m    = lane & 15;   // column / row-within-tile index
    const int half = lane >> 4;   // which 16-lane half of the wave

    const float* q = q_g + (size_t)batch * (SQ * DH);
    const float* v = v_g + (size_t)batch * (SQ * DH);
    float* lv = lds_v[wave];
    float* la = lds_a[wave];

    // ---- Stage V into LDS (coalesced b128 loads, padded rows) -------------
#pragma unroll
    for (int i = 0; i < 4; ++i) {
        int idx = i * 128 + lane * 4;          // flat float index into V[32][16]
        v4f t = *(const v4f*)(v + idx);
        int row = idx >> 4;
        int col = idx & 15;                    // multiple of 4 -> 16B aligned in LDS
        *(v4f*)(lv + row * VPAD + col) = t;
    }
    __syncthreads();

    // ---- A fragments of Q (16x4 f32 per WMMA): straight from global -------
    // lane(half,m) holds { Q[r0+m][kk*4+2*half], Q[r0+m][kk*4+2*half+1] }
    v2f aQ[2][4];
#pragma unroll
    for (int qt = 0; qt < 2; ++qt)
#pragma unroll
        for (int kk = 0; kk < 4; ++kk)
            aQ[qt][kk] = *(const v2f*)(q + (qt * 16 + m) * DH + kk * 4 + 2 * half);

    // ---- B fragments of V^T (4x16 f32 per WMMA): from LDS -----------------
    // B[k][n] = V[n][kk*4+k]; lane(half,m) -> { V[c0+m][kk*4+2h], V[c0+m][kk*4+2h+1] }
    v2f bV[2][4];
#pragma unroll
    for (int kt = 0; kt < 2; ++kt)
#pragma unroll
        for (int kk = 0; kk < 4; ++kk)
            bV[kt][kk] = *(const v2f*)(lv + (kt * 16 + m) * VPAD + kk * 4 + 2 * half);

    // ---- score = Q V^T : four 16x16 tiles, K=16 via 4x wmma 16x16x4 -------
    v8f sc[2][2];
#pragma unroll
    for (int qt = 0; qt < 2; ++qt)
#pragma unroll
        for (int kt = 0; kt < 2; ++kt) {
            v8f c = {0.f, 0.f, 0.f, 0.f, 0.f, 0.f, 0.f, 0.f};
#pragma unroll
            for (int kk = 0; kk < 4; ++kk)
                c = __builtin_amdgcn_wmma_f32_16x16x4_f32(
                        false, aQ[qt][kk], false, bV[kt][kk],
                        (short)0, c, false, false);
            sc[qt][kt] = c;
        }

    // ---- Publish raw scores to LDS in C layout ----------------------------
    // sc[qt][kt][j] at lane(half,m) = score[qt*16 + j + 8*half][kt*16 + m].
    // Halves hit disjoint 16-bank ranges (8*36 = 288 ≡ 32 mod 64): conflict-free.
#pragma unroll
    for (int qt = 0; qt < 2; ++qt)
#pragma unroll
        for (int kt = 0; kt < 2; ++kt)
#pragma unroll
            for (int j = 0; j < 8; ++j) {
                int row = qt * 16 + j + 8 * half;
                la[row * APAD + kt * 16 + m] = sc[qt][kt][j];
            }
    __syncthreads();

    // ---- Row-per-lane softmax: lane r owns score row r --------------------
    // 8 aligned b128 LDS loads (row pitch 144B -> 16B aligned), pure VALU
    // max/exp/sum/scale, zero cross-lane shuffles.
    v4f xv[8];
    {
        const float* lr = la + lane * APAD;
#pragma unroll
        for (int i = 0; i < 8; ++i)
            xv[i] = *(const v4f*)(lr + i * 4);
    }
    float mx = xv[0][0];
#pragma unroll
    for (int i = 0; i < 8; ++i)
#pragma unroll
        for (int c = 0; c < 4; ++c)
            mx = fmaxf(mx, xv[i][c]);
    float s = 0.f;
#pragma unroll
    for (int i = 0; i < 8; ++i)
#pragma unroll
        for (int c = 0; c < 4; ++c) {
            float e = __expf(xv[i][c] - mx);
            xv[i][c] = e;
            s += e;
        }
    float inv = 1.0f / s;
#pragma unroll
    for (int i = 0; i < 8; ++i)
        xv[i] *= inv;

    // Write normalized attn: own row back to LDS (A-layout source for GEMM2)
    // and straight to global (one 128B cache line per lane).
    {
        float* lw = la + lane * APAD;
        float* ga = attn_out + (size_t)batch * (SQ * SQ) + lane * SQ;
#pragma unroll
        for (int i = 0; i < 8; ++i) {
            *(v4f*)(lw + i * 4) = xv[i];
            *(v4f*)(ga + i * 4) = xv[i];
        }
    }
    __syncthreads();

    // ---- context = attn @ V : two 16x16 tiles, K=32 via 8x wmma 16x16x4 ---
    float* co = ctx_out + (size_t)batch * (SQ * DH);
#pragma unroll
    for (int qt = 0; qt < 2; ++qt) {
        v8f c = {0.f, 0.f, 0.f, 0.f, 0.f, 0.f, 0.f, 0.f};
#pragma unroll
        for (int kc = 0; kc < 8; ++kc) {
            // A frag: attn rows qt*16+m, K-cols kc*4+2*half (strided LDS, pad=36)
            v2f a = *(const v2f*)(la + (qt * 16 + m) * APAD + kc * 4 + 2 * half);
            // B frag: B[k][n] = V[kc*4+k][n]; row-broadcast LDS reads, conflict-free
            int kr = kc * 4 + 2 * half;
            v2f b;
            b.x = lv[kr * VPAD + m];
            b.y = lv[(kr + 1) * VPAD + m];
            c = __builtin_amdgcn_wmma_f32_16x16x4_f32(
                    false, a, false, b, (short)0, c, false, false);
        }
#pragma unroll
        for (int j = 0; j < 8; ++j) {
            int row = qt * 16 + j + 8 * half;
            co[row * DH + m] = c[j];       // coalesced 64B runs
        }
    }
}

extern "C" void kernel_launch(void* const* d_in, const int* in_sizes, int n_in,
                              void* d_out, int out_size, void* d_ws, size_t ws_size,
                              hipStream_t stream) {
    (void)n_in; (void)out_size; (void)d_ws; (void)ws_size;
    const float* q = (const float*)d_in[0];
    const float* v = (const float*)d_in[1];
    float* out = (float*)d_out;

    const int B = in_sizes[0] / (SQ * DH);          // 131072
    float* ctx  = out;                               // [B,32,16]
    float* attn = out + (size_t)B * SQ * DH;         // [B,1024]

    const int blocks = B / WPB;                      // B = 131072, divisible by 8
    dpa_fused_kernel<<<blocks, WPB * 32, 0, stream>>>(q, v, ctx, attn);
}